// kurtosislearn_14104672600185
// MI455X (gfx1250) — compile-verified
//
#include <hip/hip_runtime.h>
#include <hip/hip_bf16.h>
#include <stddef.h>

typedef __attribute__((ext_vector_type(2))) float v2f;
typedef __attribute__((ext_vector_type(8))) float v8f;

static constexpr int C_   = 512;
static constexpr int T_   = 8192;
static constexpr int WIN  = 16;
static constexpr int RAD  = 8;
static constexpr int TN   = T_ - WIN;      // 8176 (multiple of 16)
static constexpr int CP   = C_ + 2 * RAD;  // 528 padded channel rows

// ---------------------------------------------------------------------------
// Kernel 1: S[c,t] = sum_{i=t-8}^{t+7} inp[c,i]  (zeros outside [0,T)).
// Written into channel-padded Spad[528][TN]; rows [0,8) and [520,528) are zero
// so the WMMA kernel needs no channel bounds checks.
// ---------------------------------------------------------------------------
__global__ __launch_bounds__(256) void k_window_sum(const float* __restrict__ inp,
                                                    float* __restrict__ Spad) {
    int t   = blockIdx.x * 256 + threadIdx.x;
    int row = blockIdx.y;                      // 0..527
    if (t >= TN) return;
    float s = 0.0f;
    if (row >= RAD && row < C_ + RAD) {
        const float* r = inp + (size_t)(row - RAD) * T_;
        #pragma unroll
        for (int k = 0; k < WIN; ++k) {
            int i = t - RAD + k;               // i <= 8182 < T always
            s += (i >= 0) ? r[i] : 0.0f;
        }
    }
    Spad[(size_t)row * TN + t] = s;
}

// ---------------------------------------------------------------------------
// Kernel 2: channel-window kurtosis via V_WMMA_F32_16X16X4_F32.
// One wave per 16(ch) x 16(t) tile. Band matrix A (0/1, width-17) built in
// registers per ISA A-layout; B fragments are S^p tiles per ISA B-layout.
// 8 k-steps x 4 powers = 32 WMMAs/tile. Result transposed via LDS into
// kurtT[t][c].
// ---------------------------------------------------------------------------
__global__ __launch_bounds__(32) void k_kurt_wmma(const float* __restrict__ Spad,
                                                  float* __restrict__ kurtT) {
    __shared__ float lds[16][17];
    const int lane = threadIdx.x;        // 0..31
    const int t0   = blockIdx.x * 16;
    const int c0   = blockIdx.y * 16;
    const int n    = lane & 15;          // A: row M=n ; B: col N=n (time)
    const int hi   = lane >> 4;          // K half select

    v8f acc1 = {0,0,0,0,0,0,0,0};
    v8f acc2 = {0,0,0,0,0,0,0,0};
    v8f acc3 = {0,0,0,0,0,0,0,0};
    v8f acc4 = {0,0,0,0,0,0,0,0};

    #pragma unroll
    for (int j = 0; j < 8; ++j) {
        const int klo = j * 4 + hi * 2;  // local k index of VGPR0 element
        // A[m][k] = 1 iff |m - k| <= 8  ->  0 <= m+8-k <= 16
        const int d0 = n + 8 - klo;
        const int d1 = d0 - 1;
        v2f A;
        A.x = (d0 >= 0 && d0 <= 16) ? 1.0f : 0.0f;
        A.y = (d1 >= 0 && d1 <= 16) ? 1.0f : 0.0f;

        // B rows are padded-channel rows c0+klo, c0+klo+1 at time t0+n
        const size_t base = (size_t)(c0 + klo) * TN + (t0 + n);
        const float s0 = Spad[base];
        const float s1 = Spad[base + TN];
        v2f B1 = { s0,          s1          };
        v2f B2 = { s0 * s0,     s1 * s1     };
        v2f B3 = { B2.x * s0,   B2.y * s1   };
        v2f B4 = { B2.x * B2.x, B2.y * B2.y };

        acc1 = __builtin_amdgcn_wmma_f32_16x16x4_f32(false, A, false, B1, (short)0, acc1, false, false);
        acc2 = __builtin_amdgcn_wmma_f32_16x16x4_f32(false, A, false, B2, (short)0, acc2, false, false);
        acc3 = __builtin_amdgcn_wmma_f32_16x16x4_f32(false, A, false, B3, (short)0, acc3, false, false);
        acc4 = __builtin_amdgcn_wmma_f32_16x16x4_f32(false, A, false, B4, (short)0, acc4, false, false);
    }

    // kurtosis from raw moments; D layout: VGPR e -> M = hi*8+e, N = n
    const float inv_n = 1.0f / 17.0f;
    #pragma unroll
    for (int e = 0; e < 8; ++e) {
        const float s1 = acc1[e], s2 = acc2[e], s3 = acc3[e], s4 = acc4[e];
        const float mu  = s1 * inv_n;
        const float mu2 = mu * mu;
        const float m2  = s2 * inv_n - mu2;
        const float m4  = s4 * inv_n - 4.0f * mu * (s3 * inv_n)
                        + 6.0f * mu2 * (s2 * inv_n) - 3.0f * mu2 * mu2;
        lds[hi * 8 + e][n] = m4 / (m2 * m2) - 3.0f;
    }
    __syncthreads();

    // transpose to kurtT[t][c]; each lane writes 8 contiguous floats
    const int tr = lane >> 1;            // 0..15 (time within tile)
    const int mh = (lane & 1) * 8;       // channel half
    float* dst = kurtT + (size_t)(t0 + tr) * C_ + (c0 + mh);
    #pragma unroll
    for (int i = 0; i < 8; ++i) dst[i] = lds[mh + i][tr];
}

// ---------------------------------------------------------------------------
// Kernel 3: xT[t][c] = (t>=8) ? inp[c][t-8] : 0   (LDS tile transpose)
// ---------------------------------------------------------------------------
__global__ __launch_bounds__(256) void k_xT(const float* __restrict__ inp,
                                            float* __restrict__ xT) {
    __shared__ float lds[16][17];
    const int tx = threadIdx.x & 15, ty = threadIdx.x >> 4;
    const int t0 = blockIdx.x * 16, c0 = blockIdx.y * 16;
    const int t  = t0 + tx;
    lds[ty][tx] = (t >= RAD) ? inp[(size_t)(c0 + ty) * T_ + (t - RAD)] : 0.0f;
    __syncthreads();
    xT[(size_t)(t0 + ty) * C_ + (c0 + tx)] = lds[tx][ty];
}

// ---------------------------------------------------------------------------
// Kernel 4: per-channel sequential scan. 16 waves, 1 per block, one lane per
// channel. Batch-of-8 loads with one-batch software prefetch to keep ~16
// L2 loads in flight per iteration.
// ---------------------------------------------------------------------------
__global__ __launch_bounds__(32) void k_scan(const float* __restrict__ kurtT,
                                             const float* __restrict__ xT,
                                             const float* __restrict__ w_init,
                                             float* __restrict__ spkT) {
    const int c = blockIdx.x * 32 + threadIdx.x;
    float w = w_init[c];
    float mem = 0.0f;

    float k[8], x[8], kn[8], xn[8];
    #pragma unroll
    for (int u = 0; u < 8; ++u) {
        k[u] = kurtT[(size_t)u * C_ + c];
        x[u] = xT[(size_t)u * C_ + c];
    }
    for (int t = 0; t < TN; t += 8) {
        if (t + 8 < TN) {
            #pragma unroll
            for (int u = 0; u < 8; ++u) {
                kn[u] = kurtT[(size_t)(t + 8 + u) * C_ + c];
                xn[u] = xT[(size_t)(t + 8 + u) * C_ + c];
            }
        }
        float s[8];
        #pragma unroll
        for (int u = 0; u < 8; ++u) {
            w = fminf(fmaxf(0.5f * w + 0.5f * k[u], 0.0f), 10.0f);
            mem = 0.95f * mem + w * x[u];
            const float spk = (mem > 1.0f) ? 1.0f : 0.0f;
            mem -= spk;
            s[u] = spk;
        }
        #pragma unroll
        for (int u = 0; u < 8; ++u) spkT[(size_t)(t + u) * C_ + c] = s[u];
        #pragma unroll
        for (int u = 0; u < 8; ++u) { k[u] = kn[u]; x[u] = xn[u]; }
    }
}

// ---------------------------------------------------------------------------
// Kernel 5: out[c][t] = spkT[t][c] for t < TN  (LDS tile transpose)
// ---------------------------------------------------------------------------
__global__ __launch_bounds__(256) void k_outT(const float* __restrict__ spkT,
                                              float* __restrict__ out) {
    __shared__ float lds[16][17];
    const int tx = threadIdx.x & 15, ty = threadIdx.x >> 4;
    const int t0 = blockIdx.x * 16, c0 = blockIdx.y * 16;
    lds[ty][tx] = spkT[(size_t)(t0 + ty) * C_ + (c0 + tx)];
    __syncthreads();
    out[(size_t)(c0 + ty) * T_ + (t0 + tx)] = lds[tx][ty];
}

// ---------------------------------------------------------------------------
// Kernel 6: tail columns [TN, T) copied straight from inp.
// ---------------------------------------------------------------------------
__global__ __launch_bounds__(256) void k_tail(const float* __restrict__ inp,
                                              float* __restrict__ out) {
    const int idx = blockIdx.x * 256 + threadIdx.x;   // 0..8191
    const int c = idx >> 4;
    const int t = TN + (idx & 15);
    out[(size_t)c * T_ + t] = inp[(size_t)c * T_ + t];
}

// ---------------------------------------------------------------------------
extern "C" void kernel_launch(void* const* d_in, const int* in_sizes, int n_in,
                              void* d_out, int out_size, void* d_ws, size_t ws_size,
                              hipStream_t stream) {
    const float* inp    = (const float*)d_in[0];   // 512*8192 f32
    const float* w_init = (const float*)d_in[1];   // 512 f32
    float* out = (float*)d_out;

    float* ws    = (float*)d_ws;
    float* Spad  = ws;                                  // 528*TN
    float* kurtT = Spad  + (size_t)CP * TN;             // TN*512
    float* xT    = kurtT + (size_t)TN * C_;             // TN*512
    float* spkT  = xT    + (size_t)TN * C_;             // TN*512

    k_window_sum<<<dim3((TN + 255) / 256, CP), 256, 0, stream>>>(inp, Spad);
    k_kurt_wmma <<<dim3(TN / 16, C_ / 16),      32, 0, stream>>>(Spad, kurtT);
    k_xT        <<<dim3(TN / 16, C_ / 16),     256, 0, stream>>>(inp, xT);
    k_scan      <<<dim3(C_ / 32),               32, 0, stream>>>(kurtT, xT, w_init, spkT);
    k_outT      <<<dim3(TN / 16, C_ / 16),     256, 0, stream>>>(spkT, out);
    k_tail      <<<dim3(32),                   256, 0, stream>>>(inp, out);
}